// STGCNLayer_46428596470469
// MI455X (gfx1250) — compile-verified
//
#include <hip/hip_runtime.h>
#include <hip/hip_bf16.h>

// ---------------------------------------------------------------------------
// STGCN layer for MI455X (gfx1250, wave32).
//   K1 fuse_weights : W3[c][k]=sum_o conv_w[o,i,tap]*gcn_w[o,c], bias2
//   K2 xpad         : zero-padded x copy [8][4098][64] -> branch-free GEMM A
//   K3 xw_gemm      : xw = conv(x) @ gcn_w via V_WMMA_F32_16X16X4_F32 (K=192)
//   K4 deg_init / K5 deg_acc / K6 dis : symmetric-norm degree -> rsqrt
//   K7 self_loop    : agg = dis^2 * xw        (self-loop contribution)
//   K8 edge_agg     : agg[col] += dis[row]*w*dis[col] * xw[row]
//                     (wave-cooperative: shfl-broadcast metadata, coalesced
//                      float2 gathers, 2 global_atomic_add_f32 per lane)
//   K9/K10 stats    : per-channel sum/sumsq   (gcn_b cancels in BN)
//   K11 bn_relu     : normalize + affine + relu -> out
// ---------------------------------------------------------------------------

#define NB     8
#define SEQ    4096
#define SEQP   (SEQ + 2)            // padded sequence length
#define CH     64
#define NNODES (NB * SEQ)           // 32768
#define NPAD   (NB * SEQP * CH)     // padded element count
#define NEDGES 1048576
#define KFUSE  192                  // 3 taps * 64 in-channels
#define WT_S   194                  // LDS stride for W3 (even, spreads banks)

typedef float v2f __attribute__((ext_vector_type(2)));
typedef float v8f __attribute__((ext_vector_type(8)));

#if defined(__gfx1250__) && __has_builtin(__builtin_amdgcn_wmma_f32_16x16x4_f32)
#define HAVE_WMMA_F32 1
#else
#define HAVE_WMMA_F32 0
#endif

// --------------------------- K1: fused weights ------------------------------
__global__ __launch_bounds__(256) void stgcn_fuse_weights(
    const float* __restrict__ conv_w, const float* __restrict__ conv_b,
    const float* __restrict__ gcn_w, float* __restrict__ W3,
    float* __restrict__ bias2) {
  int t = blockIdx.x * 256 + threadIdx.x;
  if (t < CH * KFUSE) {
    int c = t / KFUSE, k = t - c * KFUSE;
    int tap = k >> 6, i = k & 63;
    float s = 0.f;
    for (int o = 0; o < CH; ++o)
      s += conv_w[o * KFUSE + i * 3 + tap] * gcn_w[o * CH + c];
    W3[c * KFUSE + k] = s;
  } else if (t < CH * KFUSE + CH) {
    int c = t - CH * KFUSE;
    float s = 0.f;
    for (int o = 0; o < CH; ++o) s += conv_b[o] * gcn_w[o * CH + c];
    bias2[c] = s;
  }
}

// --------------------------- K2: zero-padded x ------------------------------
// xpad[b][sp][c] = (1 <= sp <= 4096) ? x[b][sp-1][c] : 0
__global__ __launch_bounds__(256) void stgcn_xpad(
    const float* __restrict__ x, float* __restrict__ xpad) {
  int idx = blockIdx.x * 256 + threadIdx.x;
  if (idx < NPAD) {
    int c = idx & 63;
    int prow = idx >> 6;
    int b = prow / SEQP;
    int s = prow - b * SEQP - 1;
    float v = 0.f;
    if ((unsigned)s < (unsigned)SEQ) v = x[(long)(b * SEQ + s) * CH + c];
    xpad[idx] = v;
  }
}

// --------------------------- K3: fused GEMM (WMMA) ---------------------------
// Per wave: 16 node-rows x 64 channels, K=192 in 48 f32 WMMA steps.
// f32 16x16x4 layouts (ISA 7.12.2):
//   A: lanes0-15 -> M=lane, K={k0,k0+1}; lanes16-31 -> K={k0+2,k0+3}
//   B: VGPR pair -> K rows {k0,k0+1}/{k0+2,k0+3}, N = lane&15
//   C/D: VGPR v -> M = v + 8*(lane>=16), N = lane&15
// With xpad, the A address is linear in kc: P + 4*kc + 2*half (branch-free).
__global__ __launch_bounds__(256) void stgcn_xw_gemm(
    const float* __restrict__ xpad, const float* __restrict__ W3,
    const float* __restrict__ bias2, float* __restrict__ xw) {
  __shared__ float sWT[CH * WT_S];  // 49664 B

  const int tid = threadIdx.x;
  for (int i = tid; i < CH * KFUSE; i += 256) {
    int c = i / KFUSE, k = i - c * KFUSE;
    sWT[c * WT_S + k] = W3[i];
  }
  __syncthreads();

  const int wave = tid >> 5;
  const int lane = tid & 31;
  const int half = lane >> 4;   // 0: K pair {0,1}, 1: K pair {2,3}
  const int l15  = lane & 15;
  const int nodeBase = blockIdx.x * 128 + wave * 16;
  const int b  = nodeBase >> 12;         // batch
  const int s0 = nodeBase & (SEQ - 1);   // seq offset of first row

  // Lane's A row pointer into padded x: padded row = b*4098 + s0 + l15 (+tap)
  const float* P =
      xpad + (long)(b * SEQP + s0 + l15) * CH + 2 * half;

  v8f acc[4];
  for (int t = 0; t < 4; ++t) acc[t] = (v8f)0.0f;

#if HAVE_WMMA_F32
#pragma unroll 4
  for (int kc = 0; kc < 48; ++kc) {
    const v2f a = *(const v2f*)(P + 4 * kc);  // = row(tap), ch pair
    const int k0 = kc * 4 + half * 2;
#pragma unroll
    for (int t = 0; t < 4; ++t) {
      const v2f bb = *(const v2f*)(&sWT[(t * 16 + l15) * WT_S + k0]);
      acc[t] = __builtin_amdgcn_wmma_f32_16x16x4_f32(
          false, a, false, bb, (short)0, acc[t], false, false);
    }
  }
#else
  // Scalar fallback producing identical per-lane C/D layout.
  for (int t = 0; t < 4; ++t) {
    const int col = t * 16 + l15;
    for (int v = 0; v < 8; ++v) {
      const int m = v + half * 8;
      float s = 0.f;
      for (int k = 0; k < KFUSE; ++k) {
        const int tap = k >> 6, chv = k & 63;
        float xv = xpad[(long)(b * SEQP + s0 + m + tap) * CH + chv];
        s += xv * sWT[col * WT_S + k];
      }
      acc[t][v] = s;
    }
  }
#endif

#pragma unroll
  for (int t = 0; t < 4; ++t) {
    const int col = t * 16 + l15;
    const float b2 = bias2[col];
    float* outp = xw + (long)nodeBase * CH + col;
#pragma unroll
    for (int v = 0; v < 8; ++v) {
      const int m = v + half * 8;
      outp[m * CH] = acc[t][v] + b2;
    }
  }
}

// --------------------------- K4-K6: degrees ---------------------------------
__global__ __launch_bounds__(256) void stgcn_deg_init(float* __restrict__ deg) {
  int i = blockIdx.x * 256 + threadIdx.x;
  if (i < NNODES) deg[i] = 1.0f;  // self-loop weight
}

__global__ __launch_bounds__(256) void stgcn_deg_acc(
    const long long* __restrict__ ei, const float* __restrict__ ew,
    float* __restrict__ deg) {
  int e = blockIdx.x * 256 + threadIdx.x;
  if (e < NEDGES) {
    int col = (int)ei[NEDGES + e];
    atomicAdd(&deg[col], ew[e]);
  }
}

__global__ __launch_bounds__(256) void stgcn_dis(float* __restrict__ dis) {
  int i = blockIdx.x * 256 + threadIdx.x;
  if (i < NNODES) dis[i] = rsqrtf(dis[i]);  // deg >= 1 always
}

// --------------------------- K7: self-loop init ------------------------------
__global__ __launch_bounds__(256) void stgcn_self_loop(
    const float* __restrict__ xw, const float* __restrict__ dis,
    float* __restrict__ agg) {
  int idx = blockIdx.x * 256 + threadIdx.x;
  if (idx < NNODES * CH) {
    int n = idx >> 6;
    float d = dis[n];
    agg[idx] = d * d * xw[idx];
  }
}

// --------------------------- K8: edge aggregation ----------------------------
// Wave-cooperative: each lane loads one edge's metadata (coalesced), then a
// 32-step shfl-broadcast loop: full wave handles the 64 channels of edge j
// (float2 gather from L2-resident xw + 2 global_atomic_add_f32 per lane).
__global__ __launch_bounds__(256) void stgcn_edge_agg(
    const long long* __restrict__ ei, const float* __restrict__ ew,
    const float* __restrict__ dis, const float* __restrict__ xw,
    float* __restrict__ agg) {
  const int lane = threadIdx.x & 31;
  const int wave = threadIdx.x >> 5;
  const int eBase = blockIdx.x * 256 + wave * 32;
  const int e = eBase + lane;

  const int rowL = (int)ei[e];
  const int colL = (int)ei[NEDGES + e];
  const float coeffL = dis[rowL] * ew[e] * dis[colL];
  const int c2 = lane * 2;

  for (int j = 0; j < 32; ++j) {
    const int row = __shfl(rowL, j, 32);
    const int col = __shfl(colL, j, 32);
    const float coeff = __shfl(coeffL, j, 32);
    const v2f v = *(const v2f*)(xw + row * CH + c2);
    float* dst = agg + col * CH + c2;
    atomicAdd(dst + 0, coeff * v.x);
    atomicAdd(dst + 1, coeff * v.y);
  }
}

// --------------------------- K9-K11: BatchNorm + ReLU ------------------------
__global__ __launch_bounds__(128) void stgcn_zero_sums(float* __restrict__ s) {
  s[threadIdx.x] = 0.f;
}

__global__ __launch_bounds__(256) void stgcn_stats(
    const float* __restrict__ agg, float* __restrict__ sums) {
  __shared__ float red[512];
  const int tid = threadIdx.x;
  const int c = tid & 63, rg = tid >> 6;  // 4 row-groups
  const long rowBase = blockIdx.x * 256;  // 256 rows per block
  float s = 0.f, s2 = 0.f;
  for (int i = 0; i < 64; ++i) {
    float v = agg[(rowBase + rg + i * 4) * CH + c];
    s += v;
    s2 += v * v;
  }
  red[tid] = s;
  red[256 + tid] = s2;
  __syncthreads();
  if (tid < 64) {
    float ts = red[tid] + red[tid + 64] + red[tid + 128] + red[tid + 192];
    float t2 = red[256 + tid] + red[256 + tid + 64] + red[256 + tid + 128] +
               red[256 + tid + 192];
    atomicAdd(&sums[tid], ts);
    atomicAdd(&sums[64 + tid], t2);
  }
}

__global__ __launch_bounds__(256) void stgcn_bn_relu(
    const float* __restrict__ agg, const float* __restrict__ sums,
    const float* __restrict__ gamma, const float* __restrict__ beta,
    float* __restrict__ out) {
  int idx = blockIdx.x * 256 + threadIdx.x;
  if (idx < NNODES * CH) {
    int c = idx & 63;
    const float invN = 1.0f / (float)NNODES;
    float mean = sums[c] * invN;
    float var = sums[64 + c] * invN - mean * mean;
    float inv = rsqrtf(var + 1e-5f);
    float y = (agg[idx] - mean) * inv * gamma[c] + beta[c];
    out[idx] = fmaxf(y, 0.f);
  }
}

// ---------------------------------------------------------------------------
extern "C" void kernel_launch(void* const* d_in, const int* in_sizes, int n_in,
                              void* d_out, int out_size, void* d_ws,
                              size_t ws_size, hipStream_t stream) {
  (void)in_sizes; (void)n_in; (void)out_size; (void)ws_size;
  const float* x       = (const float*)d_in[0];
  const long long* ei  = (const long long*)d_in[1];  // int64 [2, NEDGES]
  const float* ew      = (const float*)d_in[2];
  const float* conv_w  = (const float*)d_in[3];
  const float* conv_b  = (const float*)d_in[4];
  const float* gcn_w   = (const float*)d_in[5];
  // d_in[6] = gcn_b: cancels exactly under BatchNorm mean-subtraction.
  const float* gamma   = (const float*)d_in[7];
  const float* beta    = (const float*)d_in[8];
  float* out = (float*)d_out;

  char* ws = (char*)d_ws;
  float* W3    = (float*)(ws + 0);          // 49152 B
  float* bias2 = (float*)(ws + 49152);      // 256 B
  float* sums  = (float*)(ws + 50176);      // 512 B
  float* xpad  = (float*)(ws + 65536);      // 8,392,704 B
  float* xw    = (float*)(ws + 8458240);    // 8 MB
  float* agg   = (float*)(ws + 16846848);   // 8 MB
  float* dis   = (float*)(ws + 25235456);   // 128 KB (deg then rsqrt in-place)

  stgcn_fuse_weights<<<49, 256, 0, stream>>>(conv_w, conv_b, gcn_w, W3, bias2);
  stgcn_xpad<<<(NPAD + 255) / 256, 256, 0, stream>>>(x, xpad);
  stgcn_xw_gemm<<<NNODES / 128, 256, 0, stream>>>(xpad, W3, bias2, xw);
  stgcn_deg_init<<<NNODES / 256, 256, 0, stream>>>(dis);
  stgcn_deg_acc<<<NEDGES / 256, 256, 0, stream>>>(ei, ew, dis);
  stgcn_dis<<<NNODES / 256, 256, 0, stream>>>(dis);
  stgcn_self_loop<<<NNODES * CH / 256, 256, 0, stream>>>(xw, dis, agg);
  stgcn_edge_agg<<<NEDGES / 256, 256, 0, stream>>>(ei, ew, dis, xw, agg);
  stgcn_zero_sums<<<1, 128, 0, stream>>>(sums);
  stgcn_stats<<<NNODES / 256, 256, 0, stream>>>(agg, sums);
  stgcn_bn_relu<<<NNODES * CH / 256, 256, 0, stream>>>(agg, sums, gamma, beta,
                                                       out);
}